// LinkPredictorGNN_73993696575515
// MI455X (gfx1250) — compile-verified
//
#include <hip/hip_runtime.h>
#include <hip/hip_bf16.h>
#include <math.h>

// ---------------------------------------------------------------------------
// 2-layer GAT for MI455X (gfx1250, wave32).
// Dense GEMMs: V_WMMA_F32_16X16X4_F32, weights staged in LDS (compile-time
// strides, fully unrolled K -> back-to-back WMMA + DS co-execution).
// Edge phase: 3-pass segment softmax + scatter with f32 global atomics
// (HBM/atomic bound part; wave-per-edge coalesced row gathers).
// ---------------------------------------------------------------------------

typedef __attribute__((ext_vector_type(2))) float v2f;
typedef __attribute__((ext_vector_type(8))) float v8f;

#define NEG_SLOPE 0.2f

// ---------------- float atomic max via ordered-int trick -------------------
__device__ __forceinline__ void atomicMaxFloat(float* addr, float val) {
    if (val >= 0.0f) {
        atomicMax((int*)addr, __float_as_int(val));
    } else {
        atomicMin((unsigned int*)addr, __float_as_uint(val));
    }
}

// ---------------- generic fill ---------------------------------------------
__global__ void fill_f32(float* __restrict__ p, float v, long n) {
    long i = (long)blockIdx.x * blockDim.x + threadIdx.x;
    if (i < n) p[i] = v;
}

// ---------------- WMMA f32 GEMM with LDS-resident B ------------------------
// C[M,Ntot] = A[M,K] * B[K,Ntot], row-major. M%16==0. Whole B (K*Ntot floats)
// is cooperatively staged into LDS by the 256-thread block, then each wave
// computes one 16x16 tile with V_WMMA_F32_16X16X4_F32 fed from DS loads.
template <int Ntot, int K>
__global__ __launch_bounds__(256)
void wmma_gemm_f32_lds(const float* __restrict__ A,
                       const float* __restrict__ B,
                       float* __restrict__ C, int M) {
    constexpr int BN = K * Ntot;                 // floats in B
    __shared__ float ldsB[BN];

    // cooperative float4 staging of B into LDS (BN % 1024 == 0)
    #pragma unroll
    for (int i = threadIdx.x * 4; i < BN; i += 256 * 4) {
        *(float4*)&ldsB[i] = *(const float4*)&B[i];
    }
    __syncthreads();

    constexpr int tilesN = Ntot >> 4;
    const int tilesM = M >> 4;
    const int tile = blockIdx.x * 8 + (threadIdx.x >> 5);
    if (tile >= tilesM * tilesN) return;         // whole wave exits together

    const int lane = threadIdx.x & 31;
    const int half = lane >> 4;                  // 0 | 1
    const int l    = lane & 15;

    const int tm = tile / tilesN;
    const int tn = tile - tm * tilesN;

    const float* __restrict__ arow = A + (size_t)(tm * 16 + l) * K; // row M=l
    const float* __restrict__ bcol = &ldsB[tn * 16 + l];            // col N=l

    v8f acc = {};
    #pragma unroll
    for (int k = 0; k < K; k += 4) {
        const int ka = k + 2 * half;
        v2f a, b;
        a.x = arow[ka];
        a.y = arow[ka + 1];
        b.x = bcol[ka * Ntot];                    // compile-time stride
        b.y = bcol[(ka + 1) * Ntot];
        acc = __builtin_amdgcn_wmma_f32_16x16x4_f32(
                  /*neg_a=*/false, a, /*neg_b=*/false, b,
                  /*c_mod=*/(short)0, acc, /*reuse_a=*/false, /*reuse_b=*/false);
    }

    // D layout: VGPR j, lane {half,l} -> row tm*16 + j + 8*half, col tn*16 + l
    float* __restrict__ crow = C + (size_t)(tm * 16 + 8 * half) * Ntot + tn * 16 + l;
    #pragma unroll
    for (int j = 0; j < 8; ++j) crow[(size_t)j * Ntot] = acc[j];
}

// ---------------- per-(node,head) attention coefficients -------------------
template <int H, int C>
__global__ void node_alpha(const float* __restrict__ hfeat,
                           const float* __restrict__ att_s,
                           const float* __restrict__ att_d,
                           float* __restrict__ asrc,
                           float* __restrict__ adst, int Nn) {
    int t = blockIdx.x * blockDim.x + threadIdx.x;    // t = n*H + h
    if (t >= Nn * H) return;
    const int n = t / H, h = t - n * H;
    const float* __restrict__ row = hfeat + (size_t)n * (H * C) + h * C;
    float ss = 0.f, sd = 0.f;
    #pragma unroll
    for (int c = 0; c < C; ++c) {
        const float v = row[c];
        ss += v * att_s[h * C + c];
        sd += v * att_d[h * C + c];
    }
    asrc[t] = ss;
    adst[t] = sd;
}

// ---------------- edge helpers ---------------------------------------------
__device__ __forceinline__ void edge_ids(const long long* __restrict__ srcIdx,
                                         const long long* __restrict__ dstIdx,
                                         int e, int E, int& s, int& d) {
    if (e < E) { s = (int)srcIdx[e]; d = (int)dstIdx[e]; }
    else       { s = d = e - E; }                 // appended self-loop
}

// pass 1: e = leaky_relu(asrc[src]+adst[dst]); segment max over dst
template <int H>
__global__ void edge_pass1(const long long* __restrict__ srcIdx,
                           const long long* __restrict__ dstIdx,
                           int E, int Nn,
                           const float* __restrict__ asrc,
                           const float* __restrict__ adst,
                           float* __restrict__ m,
                           float* __restrict__ ebuf) {
    const int e = blockIdx.x * blockDim.x + threadIdx.x;
    if (e >= E + Nn) return;
    int s, d; edge_ids(srcIdx, dstIdx, e, E, s, d);
    #pragma unroll
    for (int h = 0; h < H; ++h) {
        float v = asrc[s * H + h] + adst[d * H + h];
        v = v > 0.f ? v : NEG_SLOPE * v;
        ebuf[(size_t)e * H + h] = v;
        atomicMaxFloat(&m[d * H + h], v);
    }
}

// pass 2: ex = exp(e - m[dst]); segment sum over dst; ebuf <- ex
template <int H>
__global__ void edge_pass2(const long long* __restrict__ srcIdx,
                           const long long* __restrict__ dstIdx,
                           int E, int Nn,
                           const float* __restrict__ m,
                           float* __restrict__ ebuf,
                           float* __restrict__ denom) {
    const int e = blockIdx.x * blockDim.x + threadIdx.x;
    if (e >= E + Nn) return;
    int s, d; edge_ids(srcIdx, dstIdx, e, E, s, d);
    (void)s;
    #pragma unroll
    for (int h = 0; h < H; ++h) {
        const float ex = __expf(ebuf[(size_t)e * H + h] - m[d * H + h]);
        ebuf[(size_t)e * H + h] = ex;
        atomicAdd(&denom[d * H + h], ex);
    }
}

// pass 3: agg[dst] += h[src] * (ex / denom[dst]); one wave per edge,
// lanes stride the H*C channels -> coalesced row gather.
template <int H, int C>
__global__ void edge_pass3(const long long* __restrict__ srcIdx,
                           const long long* __restrict__ dstIdx,
                           int E, int Nn,
                           const float* __restrict__ hfeat,
                           const float* __restrict__ ebuf,
                           const float* __restrict__ denom,
                           float* __restrict__ agg) {
    const int wave = (int)((blockIdx.x * (long)blockDim.x + threadIdx.x) >> 5);
    const int lane = threadIdx.x & 31;
    if (wave >= E + Nn) return;
    int s, d; edge_ids(srcIdx, dstIdx, wave, E, s, d);
    constexpr int HC = H * C;
    #pragma unroll
    for (int ch = lane; ch < HC; ch += 32) {
        const int h = ch / C;
        const float alpha = ebuf[(size_t)wave * H + h] / denom[d * H + h];
        atomicAdd(&agg[(size_t)d * HC + ch],
                  hfeat[(size_t)s * HC + ch] * alpha);
    }
}

// ---------------- bias + activation ----------------------------------------
__global__ void bias_elu(const float* __restrict__ agg,
                         const float* __restrict__ b,
                         float* __restrict__ out, long n, int F) {
    long i = (long)blockIdx.x * blockDim.x + threadIdx.x;
    if (i >= n) return;
    const float v = agg[i] + b[i % F];
    out[i] = v > 0.f ? v : (__expf(v) - 1.f);
}

__global__ void bias_add(const float* __restrict__ agg,
                         const float* __restrict__ b,
                         float* __restrict__ out, long n, int F) {
    long i = (long)blockIdx.x * blockDim.x + threadIdx.x;
    if (i >= n) return;
    out[i] = agg[i] + b[i % F];
}

// ---------------------------------------------------------------------------
extern "C" void kernel_launch(void* const* d_in, const int* in_sizes, int n_in,
                              void* d_out, int out_size, void* d_ws, size_t ws_size,
                              hipStream_t stream) {
    (void)n_in; (void)out_size; (void)ws_size;

    constexpr int IN = 128, HID = 32, H1 = 4, OUT = 32;
    constexpr int F1 = H1 * HID;            // 128

    const float*     x    = (const float*)d_in[0];
    const long long* ei   = (const long long*)d_in[1];   // int64 [2,E]
    const float*     W1   = (const float*)d_in[2];
    const float*     as1  = (const float*)d_in[3];
    const float*     ad1  = (const float*)d_in[4];
    const float*     b1   = (const float*)d_in[5];
    const float*     W2   = (const float*)d_in[6];
    const float*     as2  = (const float*)d_in[7];
    const float*     ad2  = (const float*)d_in[8];
    const float*     b2   = (const float*)d_in[9];
    float*           out  = (float*)d_out;

    const int N = in_sizes[0] / IN;          // 50000
    const int E = in_sizes[1] / 2;           // 800000
    const int TE = E + N;                    // edges + self loops
    const long long* srcI = ei;
    const long long* dstI = ei + E;

    // ---- workspace layout (floats) ----
    float* ws = (float*)d_ws;
    size_t off = 0;
    float* hbuf  = ws + off; off += (size_t)N * F1;   // h1_pre, later h1
    float* agg1  = ws + off; off += (size_t)N * F1;
    float* asrc1 = ws + off; off += (size_t)N * H1;
    float* adst1 = ws + off; off += (size_t)N * H1;
    float* m1    = ws + off; off += (size_t)N * H1;
    float* den1  = ws + off; off += (size_t)N * H1;
    float* ebuf1 = ws + off; off += (size_t)TE * H1;
    float* g2    = ws + off; off += (size_t)N * OUT;
    float* agg2  = ws + off; off += (size_t)N * OUT;
    float* asrc2 = ws + off; off += (size_t)N;
    float* adst2 = ws + off; off += (size_t)N;
    float* m2    = ws + off; off += (size_t)N;
    float* den2  = ws + off; off += (size_t)N;
    float* ebuf2 = ws + off; off += (size_t)TE;

    const int TB = 256;
    auto blocks = [](long n, int tb) { return (unsigned)((n + tb - 1) / tb); };

    // =================== layer 1 ===================
    fill_f32<<<blocks((long)N * H1, TB), TB, 0, stream>>>(m1,  -3.0e38f, (long)N * H1);
    fill_f32<<<blocks((long)N * H1, TB), TB, 0, stream>>>(den1, 0.0f,    (long)N * H1);
    fill_f32<<<blocks((long)N * F1, TB), TB, 0, stream>>>(agg1, 0.0f,    (long)N * F1);

    // h1_pre = x @ W1  (M=N, Ntot=128, K=128); 8 waves / block, W1 in LDS
    {
        const int tiles = (N / 16) * (F1 / 16);
        wmma_gemm_f32_lds<F1, IN><<<blocks(tiles, 8), TB, 0, stream>>>(x, W1, hbuf, N);
    }

    node_alpha<H1, HID><<<blocks((long)N * H1, TB), TB, 0, stream>>>(
        hbuf, as1, ad1, asrc1, adst1, N);

    edge_pass1<H1><<<blocks(TE, TB), TB, 0, stream>>>(srcI, dstI, E, N,
                                                      asrc1, adst1, m1, ebuf1);
    edge_pass2<H1><<<blocks(TE, TB), TB, 0, stream>>>(srcI, dstI, E, N,
                                                      m1, ebuf1, den1);
    edge_pass3<H1, HID><<<blocks((long)TE * 32, TB), TB, 0, stream>>>(
        srcI, dstI, E, N, hbuf, ebuf1, den1, agg1);

    // h1 = elu(agg1 + b1)  (written over h1_pre; h1_pre no longer needed)
    bias_elu<<<blocks((long)N * F1, TB), TB, 0, stream>>>(agg1, b1, hbuf,
                                                          (long)N * F1, F1);

    // =================== layer 2 ===================
    fill_f32<<<blocks((long)N, TB), TB, 0, stream>>>(m2,  -3.0e38f, (long)N);
    fill_f32<<<blocks((long)N, TB), TB, 0, stream>>>(den2, 0.0f,    (long)N);
    fill_f32<<<blocks((long)N * OUT, TB), TB, 0, stream>>>(agg2, 0.0f, (long)N * OUT);

    // g2 = h1 @ W2  (M=N, Ntot=32, K=128); W2 in LDS
    {
        const int tiles = (N / 16) * (OUT / 16);
        wmma_gemm_f32_lds<OUT, F1><<<blocks(tiles, 8), TB, 0, stream>>>(hbuf, W2, g2, N);
    }

    node_alpha<1, OUT><<<blocks((long)N, TB), TB, 0, stream>>>(
        g2, as2, ad2, asrc2, adst2, N);

    edge_pass1<1><<<blocks(TE, TB), TB, 0, stream>>>(srcI, dstI, E, N,
                                                     asrc2, adst2, m2, ebuf2);
    edge_pass2<1><<<blocks(TE, TB), TB, 0, stream>>>(srcI, dstI, E, N,
                                                     m2, ebuf2, den2);
    edge_pass3<1, OUT><<<blocks((long)TE * 32, TB), TB, 0, stream>>>(
        srcI, dstI, E, N, g2, ebuf2, den2, agg2);

    // out = agg2 + b2  (mean over 1 head is identity)
    bias_add<<<blocks((long)N * OUT, TB), TB, 0, stream>>>(agg2, b2, out,
                                                           (long)N * OUT, OUT);
}